// ConvNeXtParallelMoELoRA_28492813042236
// MI455X (gfx1250) — compile-verified
//
#include <hip/hip_runtime.h>

typedef __attribute__((ext_vector_type(2))) float v2f;
typedef __attribute__((ext_vector_type(8))) float v8f;

#define N_TOK     12544
#define C_DIM     768
#define HID_DIM   3072
#define M_TILE    32
#define F_CHUNK   512
#define N_CHUNKS  6          /* 3072 / 512 */
#define XS_STRIDE 772        /* 768 + 4 pad : kills 64-bank row aliasing */
#define HS_STRIDE 516        /* 512 + 4 pad */

__device__ __forceinline__ float gelu_exact(float v) {
    return 0.5f * v * (1.0f + erff(v * 0.70710678118654752f));
}

// ---------------------------------------------------------------------------
// Fused ConvNeXt MLP:  out = GELU(x @ w1 + b1) @ w2 + b2
// One workgroup = 32 tokens, 8 waves (wave32). FP32 WMMA (16x16x4).
// Each wave streams one B fragment and feeds BOTH 16-row m-tiles with it,
// so the hot loop costs 1 global b32 + 1 ds b64 per WMMA.
// ---------------------------------------------------------------------------
__global__ __launch_bounds__(256, 1) void fused_mlp_kernel(
    const float* __restrict__ x,  const float* __restrict__ w1,
    const float* __restrict__ b1, const float* __restrict__ w2,
    const float* __restrict__ b2, float* __restrict__ out)
{
    __shared__ float xs[M_TILE * XS_STRIDE];   //  98,816 B
    __shared__ float hs[M_TILE * HS_STRIDE];   //  66,048 B

    const int tid  = threadIdx.x;
    const int wave = tid >> 5;
    const int lane = tid & 31;
    const int ln   = lane & 15;   // row/col index inside a 16-wide frag
    const int kh   = lane >> 4;   // which K-half of the 16x16x4 fragment
    const int kb   = kh * 2;      // k base for this lane half

    // ---- stage x tile [32 x 768] into LDS with b128 copies ----------------
    const float* xsrc = x + (size_t)blockIdx.x * M_TILE * C_DIM;
    #pragma unroll 4
    for (int i = tid; i < (M_TILE * C_DIM) / 4; i += 256) {
        const int flat = i * 4;
        const int row  = flat / C_DIM;
        const int col  = flat - row * C_DIM;
        *(float4*)(&xs[row * XS_STRIDE + col]) = *(const float4*)(xsrc + flat);
    }
    __syncthreads();

    v8f acc[2][6];
    #pragma unroll
    for (int mt = 0; mt < 2; ++mt)
        #pragma unroll
        for (int nt = 0; nt < 6; ++nt)
            acc[mt][nt] = (v8f)(0.0f);

    const float* arow0x = &xs[ln * XS_STRIDE];          // m-tile 0 row
    const float* arow1x = &xs[(16 + ln) * XS_STRIDE];   // m-tile 1 row
    const float* arow0h = &hs[ln * HS_STRIDE];
    const float* arow1h = &hs[(16 + ln) * HS_STRIDE];

    for (int chunk = 0; chunk < N_CHUNKS; ++chunk) {
        const int f0 = chunk * F_CHUNK;

        // gfx1250 prefetch of next chunk's weight panels (speculative, RT)
        if (chunk + 1 < N_CHUNKS) {
            const int f1 = f0 + F_CHUNK;
            __builtin_prefetch(w1 + f1 + (size_t)tid * 16, 0, 3);
            __builtin_prefetch(w2 + (size_t)f1 * C_DIM + (size_t)tid * 16, 0, 3);
        }

        // ---- Phase A: hs[32 x 512] = GELU(x_tile @ w1[:, f0:f0+512] + b1) --
        // wave owns h columns [wave*64, wave*64+64) : 4 n-tiles x 2 m-tiles
        #pragma unroll 1
        for (int t4 = 0; t4 < 4; ++t4) {
            const int nt = wave * 4 + t4;
            const float* bptr = w1 + (size_t)(f0 + nt * 16 + ln);
            v8f c0 = (v8f)(0.0f);
            v8f c1 = (v8f)(0.0f);
            #pragma unroll 4
            for (int kk = 0; kk < C_DIM; kk += 4) {
                v2f b;
                b.x = bptr[(size_t)(kk + kb)     * HID_DIM];
                b.y = bptr[(size_t)(kk + kb + 1) * HID_DIM];
                const v2f a0 = *(const v2f*)(arow0x + kk + kb);  // ds_load_b64
                const v2f a1 = *(const v2f*)(arow1x + kk + kb);
                c0 = __builtin_amdgcn_wmma_f32_16x16x4_f32(
                        false, a0, false, b, (short)0, c0, false, false);
                c1 = __builtin_amdgcn_wmma_f32_16x16x4_f32(
                        false, a1, false, b, (short)0, c1, false, false);
            }
            const float bias = b1[f0 + nt * 16 + ln];
            float* hcol = &hs[nt * 16 + ln];
            #pragma unroll
            for (int g = 0; g < 8; ++g) {
                const int r0 = g + 8 * kh;
                hcol[r0 * HS_STRIDE]        = gelu_exact(c0[g] + bias);
                hcol[(16 + r0) * HS_STRIDE] = gelu_exact(c1[g] + bias);
            }
        }
        __syncthreads();

        // ---- Phase B: acc += hs @ w2[f0:f0+512, :] -------------------------
        // wave owns output columns [wave*96, wave*96+96): 6 n-tiles x 2 m-tiles
        #pragma unroll 1
        for (int nt = 0; nt < 6; ++nt) {
            const int col0 = wave * 96 + nt * 16;
            const float* bptr = w2 + (size_t)f0 * C_DIM + col0 + ln;
            v8f c0 = acc[0][nt];
            v8f c1 = acc[1][nt];
            #pragma unroll 4
            for (int kk = 0; kk < F_CHUNK; kk += 4) {
                v2f b;
                b.x = bptr[(size_t)(kk + kb)     * C_DIM];
                b.y = bptr[(size_t)(kk + kb + 1) * C_DIM];
                const v2f a0 = *(const v2f*)(arow0h + kk + kb);
                const v2f a1 = *(const v2f*)(arow1h + kk + kb);
                c0 = __builtin_amdgcn_wmma_f32_16x16x4_f32(
                        false, a0, false, b, (short)0, c0, false, false);
                c1 = __builtin_amdgcn_wmma_f32_16x16x4_f32(
                        false, a1, false, b, (short)0, c1, false, false);
            }
            acc[0][nt] = c0;
            acc[1][nt] = c1;
        }
        __syncthreads();
    }

    // ---- epilogue: + b2, store orig ---------------------------------------
    const size_t row_base = (size_t)blockIdx.x * M_TILE;
    #pragma unroll
    for (int mt = 0; mt < 2; ++mt)
        #pragma unroll
        for (int nt = 0; nt < 6; ++nt) {
            const int   col   = wave * 96 + nt * 16 + ln;
            const float bias2 = b2[col];
            #pragma unroll
            for (int g = 0; g < 8; ++g) {
                const size_t row = row_base + mt * 16 + g + 8 * kh;
                out[row * C_DIM + col] = acc[mt][nt][g] + bias2;
            }
        }
}

// ---------------------------------------------------------------------------
// Sparse top-2 LoRA MoE:  out[n] += sum_k p[n,k] * GELU(x[n] @ wd[e]) @ wu[e]
// One wave per token; rank R=16 lives in lanes 0..15 (dup in 16..31).
// ---------------------------------------------------------------------------
__global__ __launch_bounds__(256) void moe_lora_kernel(
    const float* __restrict__ x,    const float* __restrict__ probs,
    const int*   __restrict__ idx,  const float* __restrict__ wd,
    const float* __restrict__ wu,   float* __restrict__ out)
{
    const int wave = threadIdx.x >> 5;
    const int lane = threadIdx.x & 31;
    const int n    = blockIdx.x * 8 + wave;
    if (n >= N_TOK) return;

    const float* xr = x + (size_t)n * C_DIM;
    const int r  = lane & 15;
    const int hh = lane >> 4;

    float mo[24];
    #pragma unroll
    for (int j = 0; j < 24; ++j) mo[j] = 0.0f;

    #pragma unroll 1
    for (int k = 0; k < 2; ++k) {
        const int   e = idx[n * 2 + k];
        const float p = probs[n * 2 + k];        // SCALING == 1.0
        // down[r] = sum_c x[c] * wd[e, c, r]   (split c over lane halves)
        const float* wde = wd + (size_t)e * C_DIM * 16;
        float a = 0.0f;
        const int c0 = hh * (C_DIM / 2);
        #pragma unroll 4
        for (int c = c0; c < c0 + C_DIM / 2; ++c)
            a = fmaf(xr[c], wde[(size_t)c * 16 + r], a);
        a += __shfl_xor(a, 16, 32);
        const float d = gelu_exact(a);
        // up: mo[c] += p * d[r] * wu[e, r, c]; c = lane + 32*j
        const float* wue = wu + (size_t)e * 16 * C_DIM;
        #pragma unroll 1
        for (int rr = 0; rr < 16; ++rr) {
            const float pd   = p * __shfl(d, rr, 32);
            const float* wrc = wue + (size_t)rr * C_DIM + lane;
            #pragma unroll
            for (int j = 0; j < 24; ++j)
                mo[j] = fmaf(pd, wrc[j * 32], mo[j]);
        }
    }

    float* orow = out + (size_t)n * C_DIM + lane;
    #pragma unroll
    for (int j = 0; j < 24; ++j)
        orow[j * 32] += mo[j];
}

// ---------------------------------------------------------------------------
extern "C" void kernel_launch(void* const* d_in, const int* in_sizes, int n_in,
                              void* d_out, int out_size, void* d_ws, size_t ws_size,
                              hipStream_t stream) {
    (void)in_sizes; (void)n_in; (void)out_size; (void)d_ws; (void)ws_size;
    const float* x     = (const float*)d_in[0];
    const float* probs = (const float*)d_in[1];
    const int*   idx   = (const int*)  d_in[2];
    const float* w1    = (const float*)d_in[3];
    const float* b1    = (const float*)d_in[4];
    const float* w2    = (const float*)d_in[5];
    const float* b2    = (const float*)d_in[6];
    const float* wdn   = (const float*)d_in[7];
    const float* wup   = (const float*)d_in[8];
    float* out = (float*)d_out;

    fused_mlp_kernel<<<N_TOK / M_TILE, 256, 0, stream>>>(x, w1, b1, w2, b2, out);
    moe_lora_kernel<<<N_TOK / 8, 256, 0, stream>>>(x, probs, idx, wdn, wup, out);
}